// GNFConverter_14671608283534
// MI455X (gfx1250) — compile-verified
//
#include <hip/hip_runtime.h>

typedef __attribute__((ext_vector_type(2))) float v2f;
typedef __attribute__((ext_vector_type(8))) float v8f;

#define NB 16
#define NA 128
#define NP 8192
#define NT 5
#define CLIPR 0.3f
#define NEG_LOG2E -1.44269504088896340736f

// One wave (32 lanes) handles a tile of 16 queries for one batch.
// 8 waves / 256-thread block -> 128 queries per block.
// WMMA f32 16x16x4 computes coords . query dot products (K = x,y,z,0).
__global__ __launch_bounds__(256) void gnf_kernel(const float* __restrict__ coords,
                                                  const int* __restrict__ atom_types,
                                                  const float* __restrict__ query,
                                                  float* __restrict__ out) {
    __shared__ float4 cs[NA];  // x, y, z, |c|^2
    __shared__ int cty[NA];

    const int blocksPerBatch = NP / (16 * 8);  // 64
    const int b   = blockIdx.x / blocksPerBatch;
    const int tib = blockIdx.x % blocksPerBatch;

    const int tid = threadIdx.x;
    if (tid < NA) {
        const float* cp = coords + ((size_t)b * NA + tid) * 3;
        float x = cp[0], y = cp[1], z = cp[2];
        cs[tid] = make_float4(x, y, z, x * x + y * y + z * z);
        cty[tid] = atom_types[b * NA + tid];
    }
    __syncthreads();

    const int wave   = tid >> 5;
    const int lane   = tid & 31;
    const int laneLo = lane & 15;
    const bool hi    = lane >= 16;

    const int p0 = (tib * 8 + wave) * 16;
    const int qi = p0 + laneLo;

    const float* qp = query + ((size_t)b * NP + qi) * 3;
    const float qx = qp[0], qy = qp[1], qz = qp[2];
    const float qn2 = qx * qx + qy * qy + qz * qz;

    // B operand (4x16, K x N): V0 = {K0 lanes 0-15, K2 lanes 16-31}, V1 = {K1, K3}
    v2f bop;
    bop[0] = hi ? qz : qx;
    bop[1] = hi ? 0.0f : qy;

    float4 acc[NT];
#pragma unroll
    for (int t = 0; t < NT; ++t) acc[t] = make_float4(0.f, 0.f, 0.f, 0.f);

#pragma unroll
    for (int s = 0; s < NA / 16; ++s) {
        // A operand (16x4, M x K): lane M = lane&15; lanes 0-15 carry K0/K1, 16-31 K2/K3
        const int arow = s * 16 + laneLo;
        float4 ca = cs[arow];
        v2f aop;
        aop[0] = hi ? ca.z : ca.x;
        aop[1] = hi ? 0.0f : ca.y;

        v8f czero = {};
        // D[j]: dot(coords[s*16 + j + (hi?8:0)], query[p0 + (lane&15)])
        v8f d = __builtin_amdgcn_wmma_f32_16x16x4_f32(
            /*neg_a=*/false, aop, /*neg_b=*/false, bop,
            /*c_mod=*/(short)0, czero, /*reuse_a=*/false, /*reuse_b=*/false);

        const int abase = s * 16 + (hi ? 8 : 0);
#pragma unroll
        for (int j = 0; j < 8; ++j) {
            const int a = abase + j;
            float4 cd = cs[a];     // LDS broadcast within each half-wave
            int ty = cty[a];
            float d2 = fmaxf(cd.w + qn2 - 2.0f * d[j], 0.0f);
            float dist = __builtin_amdgcn_sqrtf(d2);           // single v_sqrt_f32
            float w = __builtin_amdgcn_exp2f(dist * NEG_LOG2E); // exp(-dist), single v_exp_f32
#pragma unroll
            for (int t = 0; t < NT; ++t) {
                float wt = (ty == t) ? w : 0.0f;
                acc[t].x = fmaf(wt, cd.x, acc[t].x);
                acc[t].y = fmaf(wt, cd.y, acc[t].y);
                acc[t].z = fmaf(wt, cd.z, acc[t].z);
                acc[t].w += wt;
            }
        }
    }

    // Merge lane L (atoms M=0..7) with lane L+16 (atoms M=8..15): both get full sums
#pragma unroll
    for (int t = 0; t < NT; ++t) {
        acc[t].x += __shfl_xor(acc[t].x, 16, 32);
        acc[t].y += __shfl_xor(acc[t].y, 16, 32);
        acc[t].z += __shfl_xor(acc[t].z, 16, 32);
        acc[t].w += __shfl_xor(acc[t].w, 16, 32);
    }

    // grads = (sum w*c)/den - q ; clip to norm 0.3 ; empty type -> 0
    // lanes 0-15 write types 0..2, lanes 16-31 write types 3..4
    float* op = out + ((size_t)b * NP + qi) * NT * 3;
    const int t0 = hi ? 3 : 0;
    const int t1 = hi ? NT : 3;
    for (int t = t0; t < t1; ++t) {
        float den = acc[t].w;
        float gx = 0.f, gy = 0.f, gz = 0.f;
        if (den > 0.0f) {
            float inv = __builtin_amdgcn_rcpf(den);            // single v_rcp_f32
            gx = fmaf(acc[t].x, inv, -qx);
            gy = fmaf(acc[t].y, inv, -qy);
            gz = fmaf(acc[t].z, inv, -qz);
            float m2 = gx * gx + gy * gy + gz * gz;
            if (m2 > CLIPR * CLIPR) {
                float sc = CLIPR * __builtin_amdgcn_rsqf(m2);  // single v_rsq_f32
                gx *= sc; gy *= sc; gz *= sc;
            }
        }
        op[t * 3 + 0] = gx;
        op[t * 3 + 1] = gy;
        op[t * 3 + 2] = gz;
    }
}

extern "C" void kernel_launch(void* const* d_in, const int* in_sizes, int n_in,
                              void* d_out, int out_size, void* d_ws, size_t ws_size,
                              hipStream_t stream) {
    const float* coords     = (const float*)d_in[0];
    const int*   atom_types = (const int*)d_in[1];
    const float* query      = (const float*)d_in[2];
    float* out = (float*)d_out;

    dim3 grid(NB * (NP / 128));  // 16 batches * 64 blocks/batch = 1024
    dim3 block(256);
    gnf_kernel<<<grid, block, 0, stream>>>(coords, atom_types, query, out);
}